// VQ_86878598463523
// MI455X (gfx1250) — compile-verified
//
#include <hip/hip_runtime.h>

typedef __attribute__((ext_vector_type(2))) float v2f;
typedef __attribute__((ext_vector_type(8))) float v8f;

#define VQ_B   32
#define VQ_C   256
#define VQ_HW  1024                 // H*W
#define VQ_N   32768                // B*H*W flattened rows
#define VQ_K   1024                 // codebook size
#define VQ_D   256                  // code dim (== C)
#define PLANE  (VQ_C * VQ_HW)       // floats per batch image

#define TILE_FLOATS (16 * VQ_D)     // one 16-code B tile = 4096 floats = 16 KB

// ---------------- codebook squared norms -> d_ws ----------------
__global__ void vq_esq_kernel(const float* __restrict__ emb,
                              float* __restrict__ e_sq) {
  int k = blockIdx.x * blockDim.x + threadIdx.x;
  if (k >= VQ_K) return;
  const float4* row = (const float4*)(emb + (size_t)k * VQ_D);
  float s = 0.f;
#pragma unroll
  for (int j = 0; j < VQ_D / 4; ++j) {
    float4 v = row[j];
    s += v.x * v.x + v.y * v.y + v.z * v.z + v.w * v.w;
  }
  e_sq[k] = s;
}

// ---------------- fused distance-GEMM + argmin + gather ----------------
// Block = 128 threads = 4 wave32s; each wave owns 16 consecutive flattened
// rows, block owns 64. Codebook tiles are DMA'd global->LDS (async, double
// buffered) once per block and shared by all 4 waves.
__global__ __launch_bounds__(128) void vq_main_kernel(
    const float* __restrict__ x, const float* __restrict__ emb,
    const float* __restrict__ e_sq,
    float* __restrict__ z_e, float* __restrict__ z_q) {
  __shared__ float btile[2 * TILE_FLOATS];  // 2 x 16KB double buffer
  __shared__ int   s_idx[64];

  const int lane = threadIdx.x & 31;
  const int wave = threadIdx.x >> 5;
  const int r    = lane & 15;   // A row within tile / result column
  const int hf   = lane >> 4;   // half-wave selects K pair {0,1} vs {2,3}

  const int row0_blk = blockIdx.x * 64;
  const int row0 = row0_blk + wave * 16;
  const int b  = row0 >> 10;    // batch (16 rows never cross a batch plane)
  const int p0 = row0 & 1023;   // h*W + w base (16-aligned)

  // Low 32 bits of a generic pointer to LDS == LDS byte address (ISA 10.2).
  const unsigned lds_base = (unsigned)(size_t)(&btile[0]);

  // ---- async DMA of one 16-code tile (16KB, contiguous rows of emb) ----
  // 128 threads x 8 x b128 = 16KB; tracked on ASYNCcnt, no VGPR staging.
  auto issue_tile = [&](int tile, int buf) {
    const char* gsrc = (const char*)(emb + (size_t)tile * TILE_FLOATS) +
                       threadIdx.x * 16u;
    unsigned lbase = lds_base + (unsigned)buf * (TILE_FLOATS * 4u) +
                     threadIdx.x * 16u;
#pragma unroll
    for (int s = 0; s < 8; ++s) {
      unsigned           loff = lbase + (unsigned)(s << 11);  // 128*16B stride
      unsigned long long ga   = (unsigned long long)(gsrc + (s << 11));
      asm volatile("global_load_async_to_lds_b128 %0, %1, off"
                   :: "v"(loff), "v"(ga) : "memory");
    }
  };

  // ---- A fragments: 16 rows x 256 channels of x, resident in VGPRs ----
  // x is (B,C,H,W): x_flat[n,c] at b*PLANE + c*HW + p (p fastest -> the 16
  // lanes of a half-wave read 16 consecutive floats: coalesced).
  const float* xp = x + (size_t)b * PLANE + (size_t)(p0 + r);
  v2f a_frag[64];
#pragma unroll
  for (int j = 0; j < 64; ++j) {
    const int c = 4 * j + 2 * hf;
    a_frag[j].x = xp[(size_t)c * VQ_HW];
    a_frag[j].y = xp[(size_t)(c + 1) * VQ_HW];
  }

  float minval[8];
  int   minidx[8];
#pragma unroll
  for (int m = 0; m < 8; ++m) { minval[m] = __builtin_inff(); minidx[m] = 0; }

  issue_tile(0, 0);  // prologue: tile 0 -> buf 0

  // ---- sweep the 64 codebook tiles ----
  for (int t = 0; t < 64; ++t) {
    const int buf = t & 1;
    __syncthreads();                 // everyone done reading buf^1 (tile t-1)
    if (t + 1 < 64) {
      issue_tile(t + 1, buf ^ 1);    // prefetch next tile into other buffer
      asm volatile("s_wait_asynccnt 0x8" ::: "memory");  // tile t's 8 done
    } else {
      asm volatile("s_wait_asynccnt 0x0" ::: "memory");
    }
    __syncthreads();                 // all waves' tile-t DMA complete

    // B fragments from LDS: code row = r, channels 4j+2hf -> ds_load_b64
    const v2f* brow = (const v2f*)(btile + buf * TILE_FLOATS +
                                   ((r << 8) + 2 * hf));
    v8f acc0 = {};
    v8f acc1 = {};
#pragma unroll
    for (int j = 0; j < 32; ++j) {   // two independent WMMA chains (ILP)
      acc0 = __builtin_amdgcn_wmma_f32_16x16x4_f32(
          false, a_frag[j], false, brow[2 * j], (short)0, acc0, false, false);
      acc1 = __builtin_amdgcn_wmma_f32_16x16x4_f32(
          false, a_frag[j + 32], false, brow[2 * (j + 32)], (short)0, acc1,
          false, false);
    }

    // Lane holds column k = 16t + r for rows M = hf*8 + m.
    const float es = e_sq[16 * t + r];
#pragma unroll
    for (int m = 0; m < 8; ++m) {
      float d = es - 2.0f * (acc0[m] + acc1[m]);
      if (d < minval[m]) { minval[m] = d; minidx[m] = 16 * t + r; }  // first-min
    }
  }

  // ---- reduce the 16 columns held by each half-wave (tie -> lower index) ----
#pragma unroll
  for (int m = 0; m < 8; ++m) {
    float v = minval[m];
    int   i = minidx[m];
#pragma unroll
    for (int off = 8; off >= 1; off >>= 1) {
      float ov = __shfl_xor(v, off, 16);
      int   oi = __shfl_xor(i, off, 16);
      if (ov < v || (ov == v && oi < i)) { v = ov; i = oi; }
    }
    minidx[m] = i;
  }
  if (r == 0) {
#pragma unroll
    for (int m = 0; m < 8; ++m) s_idx[wave * 16 + hf * 8 + m] = minidx[m];
  }
  __syncthreads();

  // ---- gather codes and write z_q and z_e (= z_q + (x - z_q)) ----
  const int bb = row0_blk >> 10;
  const int pb = row0_blk & 1023;          // 64-aligned
  for (int e = threadIdx.x; e < 64 * VQ_C; e += 128) {
    const int c   = e >> 6;
    const int pl  = e & 63;
    const int idx = s_idx[pl];
    const size_t off = (size_t)bb * PLANE + (size_t)c * VQ_HW + (size_t)(pb + pl);
    const float zq = emb[(size_t)idx * VQ_D + c];
    const float xv = x[off];
    z_e[off] = zq + (xv - zq);             // straight-through forward value
    z_q[off] = zq;
  }
}

extern "C" void kernel_launch(void* const* d_in, const int* in_sizes, int n_in,
                              void* d_out, int out_size, void* d_ws, size_t ws_size,
                              hipStream_t stream) {
  const float* x   = (const float*)d_in[0];   // (32,256,32,32) fp32
  const float* emb = (const float*)d_in[1];   // (1024,256)    fp32
  float* z_e  = (float*)d_out;                              // output 0
  float* z_q  = (float*)d_out + (size_t)VQ_B * PLANE;       // output 1
  float* e_sq = (float*)d_ws;                               // 1024 floats

  vq_esq_kernel<<<VQ_K / 256, 256, 0, stream>>>(emb, e_sq);
  vq_main_kernel<<<VQ_N / 64, 128, 0, stream>>>(x, emb, e_sq, z_e, z_q);
}